// fUpModule_4698694222682
// MI455X (gfx1250) — compile-verified
//
#include <hip/hip_runtime.h>

// ---------------------------------------------------------------------------
// Static problem metadata (TRIPLES structure), computed at compile time.
// ---------------------------------------------------------------------------
constexpr int MAXL_C  = 5;
constexpr int NTAU    = 24;
constexpr int BATCH_N = 128;

struct Meta {
  int ntri;
  int l1[80], l2[80], l[80];
  int cnt_l[6];
  int tri_of_l[6][24];
  int jl1[6][24], jl2[6][24];   // flattened per-l triple metadata (avoid indirection)
  int MT[6];        // middle taus per l
  int ms_off[6];    // middle-channel offset per l
  int in_off[6];    // activation element offset of F_l block (complex units)
  int w_off[6];     // weight element offset per l (complex units)
  int ms_len, in_len, w_len;
};

constexpr Meta makeMeta() {
  Meta m{};
  m.ntri = 0;
  for (int q = 0; q < 6; q++) { m.cnt_l[q] = 0; m.MT[q] = 0; }
  for (int a = 0; a <= MAXL_C; a++)
    for (int b = 0; b <= a; b++) {
      int lmax = (a + b < MAXL_C) ? (a + b) : MAXL_C;
      for (int c = a - b; c <= lmax; c++) {
        int j = m.ntri;
        m.l1[j] = a; m.l2[j] = b; m.l[j] = c;
        m.tri_of_l[c][m.cnt_l[c]] = j;
        m.jl1[c][m.cnt_l[c]] = a;
        m.jl2[c][m.cnt_l[c]] = b;
        m.cnt_l[c]++;
        m.MT[c] += NTAU * NTAU;
        m.ntri++;
      }
    }
  int acc = 0;
  for (int q = 0; q < 6; q++) { m.ms_off[q] = acc; acc += m.MT[q]; }
  m.ms_len = acc;
  acc = 0;
  for (int q = 0; q < 6; q++) { m.in_off[q] = acc; acc += NTAU * (2 * q + 1); }
  m.in_len = acc;
  acc = 0;
  for (int q = 0; q < 6; q++) { m.w_off[q] = acc; acc += NTAU * m.MT[q]; }
  m.w_len = acc;
  return m;
}

constexpr Meta kM = makeMeta();                 // compile-time constants (both sides)
__constant__ Meta dM = makeMeta();              // device-resident tables (runtime-indexed)

constexpr int IN_LEN  = kM.in_len;              // 864
constexpr int OUT_LEN = kM.in_len;              // 864
constexpr int MS_LEN  = kM.ms_len;              // 39744

// ---------------------------------------------------------------------------
// Workspace layout (all deterministic, rebuilt every launch; < 0.5 MB)
// ---------------------------------------------------------------------------
struct CgEnt { int xy; float c; };              // x | (y<<8), coefficient
constexpr int    WS_ENTS_CAP = 8192;            // actual nnz < 2.2k
constexpr size_t WS_ENTS  = 0;
constexpr size_t WS_EOFF  = WS_ENTS + WS_ENTS_CAP * sizeof(CgEnt);        // 64 KB
constexpr size_t WS_SUMSQ = WS_EOFF + 4096;                                // eoff: 69*12+1 ints
constexpr size_t WS_INV   = WS_SUMSQ + ((size_t)MS_LEN * 4 + 255) / 256 * 256;

// ---------------------------------------------------------------------------
// Kernel 1: build Clebsch-Gordan table (exact double factorials, <=16!)
// ---------------------------------------------------------------------------
__device__ double dfact(int n) { double r = 1.0; for (int i = 2; i <= n; i++) r *= (double)i; return r; }

__device__ double cg_scalar(int j1, int m1, int j2, int m2, int j, int m) {
  if (m1 + m2 != m) return 0.0;
  double pref = (double)(2 * j + 1) * dfact(j + j1 - j2) * dfact(j - j1 + j2) *
                dfact(j1 + j2 - j) / dfact(j1 + j2 + j + 1);
  pref = sqrt(pref);
  pref *= sqrt(dfact(j + m) * dfact(j - m) * dfact(j1 - m1) * dfact(j1 + m1) *
               dfact(j2 - m2) * dfact(j2 + m2));
  int kmin = 0;
  if (-(j - j2 + m1) > kmin) kmin = -(j - j2 + m1);
  if (-(j - j1 - m2) > kmin) kmin = -(j - j1 - m2);
  int kmax = j1 + j2 - j;
  if (j1 - m1 < kmax) kmax = j1 - m1;
  if (j2 + m2 < kmax) kmax = j2 + m2;
  double s = 0.0;
  for (int k = kmin; k <= kmax; k++) {
    double d = dfact(k) * dfact(j1 + j2 - j - k) * dfact(j1 - m1 - k) *
               dfact(j2 + m2 - k) * dfact(j - j2 + m1 + k) * dfact(j - j1 - m2 + k);
    s += ((k & 1) ? -1.0 : 1.0) / d;
  }
  return pref * s;
}

__global__ void build_cg(CgEnt* __restrict__ ents, int* __restrict__ eoff) {
  if (blockIdx.x != 0 || threadIdx.x != 0) return;
  int cnt = 0;
  for (int j = 0; j < dM.ntri; j++) {
    int a = dM.l1[j], b = dM.l2[j], l = dM.l[j];
    for (int mi = 0; mi < 12; mi++) {
      eoff[j * 12 + mi] = cnt;
      if (mi <= 2 * l) {
        for (int x = 0; x < 2 * a + 1; x++) {
          int y = mi - l - (x - a) + b;
          if (y >= 0 && y < 2 * b + 1) {
            double c = cg_scalar(a, x - a, b, y - b, l, mi - l);
            if (c != 0.0 && cnt < WS_ENTS_CAP) {
              ents[cnt].xy = x | (y << 8);
              ents[cnt].c = (float)c;
              cnt++;
            }
          }
        }
      }
    }
  }
  eoff[dM.ntri * 12] = cnt;
}

// ---------------------------------------------------------------------------
// Kernel 2: per-middle-channel sum of |M|^2 over (batch, m).
// One thread per channel, fixed summation order -> bit-deterministic.
// All activation data is L2-resident (0.88 MB), so re-loads are cheap.
// ---------------------------------------------------------------------------
__global__ void pass_sumsq(const float* __restrict__ act,
                           const CgEnt* __restrict__ ents,
                           const int* __restrict__ eoff,
                           float* __restrict__ sumsq) {
  int c = blockIdx.x * blockDim.x + threadIdx.x;
  if (c >= MS_LEN) return;
  int l = 5;
  for (int q = 0; q < 6; q++)
    if (c < dM.ms_off[q] + dM.MT[q]) { l = q; break; }
  int cl = c - dM.ms_off[l];
  int ktri = cl / (NTAU * NTAU);
  int j  = dM.tri_of_l[l][ktri];
  int a  = dM.jl1[l][ktri];
  int b2 = dM.jl2[l][ktri];
  int rem = cl % (NTAU * NTAU);
  int t = rem / NTAU, s = rem % NTAU;
  int d1 = 2 * a + 1, d2 = 2 * b2 + 1, dl = 2 * l + 1;
  const float* base1 = act + 2 * (dM.in_off[a]  + t * d1);
  const float* base2 = act + 2 * (dM.in_off[b2] + s * d2);
  float acc = 0.f;
  for (int mi = 0; mi < dl; mi++) {
    int e0 = eoff[j * 12 + mi], e1 = eoff[j * 12 + mi + 1];
    for (int bt = 0; bt < BATCH_N; bt++) {
      const float* f1 = base1 + 2 * bt * IN_LEN;
      const float* f2 = base2 + 2 * bt * IN_LEN;
      float sr = 0.f, si = 0.f;
      for (int e = e0; e < e1; e++) {
        CgEnt en = ents[e];
        int x = en.xy & 255, y = en.xy >> 8;
        float ar = f1[2 * x], ai = f1[2 * x + 1];
        float br = f2[2 * y], bi = f2[2 * y + 1];
        sr += en.c * (ar * br - ai * bi);
        si += en.c * (ar * bi + ai * br);
      }
      acc += sr * sr + si * si;
    }
  }
  sumsq[c] = acc;
}

// ---------------------------------------------------------------------------
// Kernel 3: fold batch-norm into a per-channel reciprocal scale.
// new_std = (ms*BM_CNT + batch_std)/(BM_CNT+1);  inv = 1/(new_std + EPS)
// ---------------------------------------------------------------------------
__global__ void pass_invstd(const float* __restrict__ sumsq,
                            const float* __restrict__ ms,
                            float* __restrict__ inv) {
  int c = blockIdx.x * blockDim.x + threadIdx.x;
  if (c >= MS_LEN) return;
  int l = 5;
  for (int q = 0; q < 6; q++)
    if (c < dM.ms_off[q] + dM.MT[q]) { l = q; break; }
  float dim = (float)(2 * l + 1);
  float bstd = sqrtf(sumsq[c] / ((float)BATCH_N * dim));
  float news = (ms[c] * 1.0f + bstd) * 0.5f;   // BM_CNT = 1.0
  inv[c] = 1.0f / (news + 1e-5f);
}

// ---------------------------------------------------------------------------
// Kernel 4 (templated on L): per-l complex mixing GEMM, V_WMMA_F32_16X16X4_F32.
//   out[b,o,m] = sum_c W[o,c] * inv_std[c] * M[b,c,m]   (complex)
// All divisors/offsets are compile-time constants -> mul-shift, no div loops.
// A = W (16 rows o x K), B = normalized M' tile (K x 16 cols (b,m)),
// M' recomputed on the fly into LDS (never materialize the 269 MB middle).
// Workgroup = 64 threads = 2 waves: wave0 -> o rows 0..15, wave1 -> 16..23.
// K chunks of 32 (MT % 32 == 0, so no tail; EXEC stays all-ones for WMMA).
// ---------------------------------------------------------------------------
typedef __attribute__((ext_vector_type(2))) float v2f;
typedef __attribute__((ext_vector_type(8))) float v8f;

template <int L>
__launch_bounds__(64)
__global__ void pass_gemm(const float* __restrict__ act,
                          const float* __restrict__ wts,
                          const CgEnt* __restrict__ ents,
                          const int* __restrict__ eoff,
                          const float* __restrict__ inv,
                          float* __restrict__ out) {
  constexpr int DIM   = 2 * L + 1;
  constexpr int MT    = kM.MT[L];
  constexpr int MSOFF = kM.ms_off[L];
  constexpr int OOFF  = kM.in_off[L];       // output block offset (same taus)
  constexpr int WOFF  = kM.w_off[L];

  __shared__ float shR[32 * 16];
  __shared__ float shI[32 * 16];

  int colBase = blockIdx.x * 16;            // grid = 8*DIM column tiles

  int wave = threadIdx.x >> 5;
  int lane = threadIdx.x & 31;
  int hi = lane >> 4, lo = lane & 15;
  int oBase = wave ? 16 : 0;

  v8f accR = {};
  v8f accI = {};
  const float* wbase = wts + 2 * (size_t)WOFF;

  for (int k0 = 0; k0 < MT; k0 += 32) {
    // ---- build normalized M' chunk (32 K-rows x 16 cols) into LDS ----
    for (int e = 0; e < 8; e++) {
      int idx = threadIdx.x + 64 * e;          // 0..511
      int kk = idx >> 4, col = idx & 15;
      int c = k0 + kk;
      int ktri = c / (NTAU * NTAU);            // constant divisor
      int j  = dM.tri_of_l[L][ktri];
      int a  = dM.jl1[L][ktri];
      int b2 = dM.jl2[L][ktri];
      int rem = c - ktri * (NTAU * NTAU);
      int t = rem / NTAU, s = rem - (rem / NTAU) * NTAU;
      int gcol = colBase + col;
      int bt = gcol / DIM, m = gcol - bt * DIM;  // constant divisor
      const float* f1 = act + 2 * (bt * IN_LEN + dM.in_off[a]  + t * (2 * a + 1));
      const float* f2 = act + 2 * (bt * IN_LEN + dM.in_off[b2] + s * (2 * b2 + 1));
      int e0 = eoff[j * 12 + m], e1 = eoff[j * 12 + m + 1];
      float sr = 0.f, si = 0.f;
      for (int q = e0; q < e1; q++) {
        CgEnt en = ents[q];
        int x = en.xy & 255, y = en.xy >> 8;
        float ar = f1[2 * x], ai = f1[2 * x + 1];
        float br = f2[2 * y], bi = f2[2 * y + 1];
        sr += en.c * (ar * br - ai * bi);
        si += en.c * (ar * bi + ai * br);
      }
      float iv = inv[MSOFF + c];
      // swizzle so the reader's (kq,hi) pair is an adjacent float2:
      // kk = kq*4 + 2*hi + jbit  ->  lidx = (kq*2+hi)*32 + col*2 + jbit
      int lidx = ((kk >> 2) * 2 + ((kk >> 1) & 1)) * 32 + col * 2 + (kk & 1);
      shR[lidx] = sr * iv;
      shI[lidx] = si * iv;
    }
    __syncthreads();

    // ---- WMMA: 8 steps of K=4; complex = 4 real f32 WMMAs per step ----
    int oA = oBase + lo; if (oA > 23) oA = 23;   // row-clamped A loads (masked on store)
    for (int kq = 0; kq < 8; kq++) {
      int cA = k0 + kq * 4 + 2 * hi;
      // weights interleaved re/im: one float4 = {Wr(c),Wi(c),Wr(c+1),Wi(c+1)}
      const float4 wv = *(const float4*)(wbase + 2 * ((size_t)oA * MT + cA));
      v2f aR; aR.x = wv.x; aR.y = wv.z;
      v2f aI; aI.x = wv.y; aI.y = wv.w;
      v2f aIn; aIn.x = -aI.x; aIn.y = -aI.y;     // f32 WMMA NEG is C-only -> negate in VALU
      const float2 bRv = *(const float2*)&shR[(kq * 2 + hi) * 32 + lo * 2];
      const float2 bIv = *(const float2*)&shI[(kq * 2 + hi) * 32 + lo * 2];
      v2f bR; bR.x = bRv.x; bR.y = bRv.y;
      v2f bI; bI.x = bIv.x; bI.y = bIv.y;
      accR = __builtin_amdgcn_wmma_f32_16x16x4_f32(false, aR,  false, bR, (short)0, accR, false, false);
      accR = __builtin_amdgcn_wmma_f32_16x16x4_f32(false, aIn, false, bI, (short)0, accR, false, false);
      accI = __builtin_amdgcn_wmma_f32_16x16x4_f32(false, aR,  false, bI, (short)0, accI, false, false);
      accI = __builtin_amdgcn_wmma_f32_16x16x4_f32(false, aI,  false, bR, (short)0, accI, false, false);
    }
    __syncthreads();
  }

  // ---- store: D layout row = r + 8*hi, col = lo ----
  for (int r = 0; r < 8; r++) {
    int o = oBase + r + 8 * hi;
    if (o >= 24) continue;
    int gcol = colBase + lo;
    int bt = gcol / DIM, m = gcol - bt * DIM;
    float* op = out + 2 * ((size_t)bt * OUT_LEN + OOFF + o * DIM + m);
    op[0] = accR[r];
    op[1] = accI[r];
  }
}

// ---------------------------------------------------------------------------
// Launch
// ---------------------------------------------------------------------------
extern "C" void kernel_launch(void* const* d_in, const int* in_sizes, int n_in,
                              void* d_out, int out_size, void* d_ws, size_t ws_size,
                              hipStream_t stream) {
  const float* act = (const float*)d_in[0];   // [128][864][2] f32
  const float* wts = (const float*)d_in[1];   // [W_LEN][2]    f32
  const float* ms  = (const float*)d_in[2];   // [MS_LEN]      f32
  float* out = (float*)d_out;                 // [128][864][2] f32

  char* ws = (char*)d_ws;
  CgEnt* ents  = (CgEnt*)(ws + WS_ENTS);
  int*   eoff  = (int*)  (ws + WS_EOFF);
  float* sumsq = (float*)(ws + WS_SUMSQ);
  float* invs  = (float*)(ws + WS_INV);

  build_cg<<<1, 64, 0, stream>>>(ents, eoff);
  pass_sumsq<<<(MS_LEN + 255) / 256, 256, 0, stream>>>(act, ents, eoff, sumsq);
  pass_invstd<<<(MS_LEN + 255) / 256, 256, 0, stream>>>(sumsq, ms, invs);

  pass_gemm<0><<< 8 * 1, 64, 0, stream>>>(act, wts, ents, eoff, invs, out);
  pass_gemm<1><<< 8 * 3, 64, 0, stream>>>(act, wts, ents, eoff, invs, out);
  pass_gemm<2><<< 8 * 5, 64, 0, stream>>>(act, wts, ents, eoff, invs, out);
  pass_gemm<3><<< 8 * 7, 64, 0, stream>>>(act, wts, ents, eoff, invs, out);
  pass_gemm<4><<< 8 * 9, 64, 0, stream>>>(act, wts, ents, eoff, invs, out);
  pass_gemm<5><<< 8 * 11, 64, 0, stream>>>(act, wts, ents, eoff, invs, out);
}